// ChamferBoundarySDFLoss_55637006352783
// MI455X (gfx1250) — compile-verified
//
#include <hip/hip_runtime.h>
#include <hip/hip_bf16.h>

// ---------------------------------------------------------------------------
// ChamferBoundarySDFLoss for MI455X (gfx1250, wave32, WMMA)
// d_in[0] = pred_logits (4,1,1024,1024) f32 ; d_in[1] = gt_sdf (4,1,1024,1024) f32
// d_out   = [loss (1 f32)] ++ [dsdf (4*1024*1024 f32)]
// ---------------------------------------------------------------------------

#define HH 1024
#define WW 1024
#define BB 4
#define KDIR 2048            // K per direction (MAX_POINTS/2)
#define PN 4096              // points per image (2*KDIR)
#define EPSF 1e-8f

typedef __attribute__((ext_vector_type(2))) float v2f;
typedef __attribute__((ext_vector_type(8))) float v8f;

// ---------------------------------------------------------------- zero d_out
__global__ void zero_out_kernel(float* __restrict__ out, int n) {
  int i = blockIdx.x * blockDim.x + threadIdx.x;
  if (i < n) out[i] = 0.0f;
}

// ------------------------------------------------------------- loss stage 1
__global__ __launch_bounds__(256)
void loss_partial_kernel(const float* __restrict__ a, const float* __restrict__ b,
                         float* __restrict__ part, int n) {
  __shared__ float sm[8];
  const int tid = threadIdx.x;
  float s = 0.0f;
  int base = blockIdx.x * (256 * 4) + tid;
#pragma unroll
  for (int k = 0; k < 4; ++k) {
    int i = base + k * 256;
    if (i < n) s += fabsf(a[i] - b[i]);
  }
#pragma unroll
  for (int off = 16; off > 0; off >>= 1) s += __shfl_down(s, off, 32);
  if ((tid & 31) == 0) sm[tid >> 5] = s;
  __syncthreads();
  if (tid == 0) {
    float t = 0.0f;
#pragma unroll
    for (int w = 0; w < 8; ++w) t += sm[w];
    part[blockIdx.x] = t;
  }
}

// ------------------------------------------------------------- loss stage 2
__global__ __launch_bounds__(1024)
void loss_final_kernel(const float* __restrict__ part, int nblk,
                       float* __restrict__ out0, float scale) {
  __shared__ float sm[32];
  const int tid = threadIdx.x;
  float s = 0.0f;
  for (int i = tid; i < nblk; i += 1024) s += part[i];
#pragma unroll
  for (int off = 16; off > 0; off >>= 1) s += __shfl_down(s, off, 32);
  if ((tid & 31) == 0) sm[tid >> 5] = s;
  __syncthreads();
  if (tid == 0) {
    float t = 0.0f;
#pragma unroll
    for (int w = 0; w < 32; ++w) t += sm[w];
    out0[0] = t * scale;
  }
}

// --------------------------------------------- ordered zero-crossing extract
// one 1024-thread block per (image b = blockIdx.x, direction = blockIdx.y)
// direction 0 = vertical (rows), 1 = horizontal (cols); first KDIR crossings
// in raster order, remaining slots filled with `sentinel`.
__global__ __launch_bounds__(1024)
void zc_extract_kernel(const float* __restrict__ sdf_all,
                       float* __restrict__ out_r, float* __restrict__ out_c,
                       float sentinel) {
  const int b = blockIdx.x;
  const int dir = blockIdx.y;
  const float* sdf = sdf_all + (size_t)b * HH * WW;
  float* orow = out_r + (size_t)b * PN + dir * KDIR;
  float* ocol = out_c + (size_t)b * PN + dir * KDIR;
  const int N = (dir == 0) ? (HH - 1) * WW : HH * (WW - 1);

  __shared__ int wsum[32];
  __shared__ int woff[32];
  __shared__ int sbase;
  __shared__ int stot;

  const int tid = threadIdx.x;
  const int lane = tid & 31;
  const int wv = tid >> 5;
  if (tid == 0) sbase = 0;
  __syncthreads();

  for (int start = 0; start < N; start += 1024) {
    if (sbase >= KDIR) break;  // uniform: sbase read after barrier
    int i = start + tid;
    bool pred = false;
    float rr = 0.0f, cc = 0.0f;
    if (i < N) {
      if (dir == 0) {
        int r = i / WW, c = i % WW;
        float v1 = sdf[r * WW + c], v2 = sdf[(r + 1) * WW + c];
        if (v1 * v2 < 0.0f) {
          pred = true;
          float av = fabsf(v1), bv = fabsf(v2);
          rr = (float)r + av / (av + bv + EPSF);
          cc = (float)c;
        }
      } else {
        int r = i / (WW - 1), c = i % (WW - 1);
        float v1 = sdf[r * WW + c], v2 = sdf[r * WW + c + 1];
        if (v1 * v2 < 0.0f) {
          pred = true;
          float av = fabsf(v1), bv = fabsf(v2);
          rr = (float)r;
          cc = (float)c + av / (av + bv + EPSF);
        }
      }
    }
    unsigned long long m = __ballot(pred);
    int prefix = __popcll(m & ((1ull << lane) - 1ull));
    int cnt = (int)__popcll(m);
    if (lane == 0) wsum[wv] = cnt;
    __syncthreads();
    if (tid == 0) {
      int acc = 0;
      for (int w = 0; w < 32; ++w) { woff[w] = acc; acc += wsum[w]; }
      stot = acc;
    }
    __syncthreads();
    if (pred) {
      int gi = sbase + woff[wv] + prefix;
      if (gi < KDIR) { orow[gi] = rr; ocol[gi] = cc; }
    }
    __syncthreads();
    if (tid == 0) sbase += stot;
    __syncthreads();
  }
  int cntv = sbase < KDIR ? sbase : KDIR;
  for (int j = cntv + tid; j < KDIR; j += 1024) {
    orow[j] = sentinel;
    ocol[j] = sentinel;
  }
}

// ------------------------------------------------- chamfer NN via f32 WMMA
// d2(p,g) = |p|^2 - 2 p.g + |g|^2.  WMMA computes D = -2 p.g + |g|^2 via
// A(16x4) = [-2pr, -2pc, 1, 0] per row, B(4x16) = [gr; gc; |g|^2; 0] per col.
// |p|^2 is a row constant: min/argmin over g unaffected; added back at the end.
// One wave per block: blockIdx.x = p-row tile (16 rows), blockIdx.y = image.
__global__ __launch_bounds__(32)
void nn_wmma_kernel(const float* __restrict__ p_r, const float* __restrict__ p_c,
                    const float* __restrict__ g_r, const float* __restrict__ g_c,
                    float* __restrict__ min_d2, int* __restrict__ nn_idx) {
  const int b = blockIdx.y;
  const int m0 = blockIdx.x * 16;
  const int lane = threadIdx.x;
  const float* pr = p_r + (size_t)b * PN;
  const float* pc = p_c + (size_t)b * PN;
  const float* gr = g_r + (size_t)b * PN;
  const float* gc = g_c + (size_t)b * PN;

  // A-matrix 16x4 f32 layout: lanes 0-15 hold (K0,K1) for M=lane;
  // lanes 16-31 hold (K2,K3) for M=lane-16.
  v2f A;
  if (lane < 16) {
    float x = pr[m0 + lane], y = pc[m0 + lane];
    A.x = -2.0f * x;
    A.y = -2.0f * y;
  } else {
    A.x = 1.0f;  // K=2 row multiplies |g|^2
    A.y = 0.0f;  // K=3 unused
  }

  float runMin[8];
  int runIdx[8];
#pragma unroll
  for (int v = 0; v < 8; ++v) { runMin[v] = 3.4e38f; runIdx[v] = 0; }

  const int nsel = lane & 15;
  for (int t = 0; t < PN; t += 16) {
    float x = gr[t + nsel], y = gc[t + nsel];
    // B-matrix 4x16 f32 layout mirrors A: lanes 0-15 = rows K0,K1 (N=lane);
    // lanes 16-31 = rows K2,K3 (N=lane-16).
    v2f Bv;
    if (lane < 16) { Bv.x = x; Bv.y = y; }
    else           { Bv.x = x * x + y * y; Bv.y = 0.0f; }
    v8f Cz = {};
    v8f D = __builtin_amdgcn_wmma_f32_16x16x4_f32(
        /*neg_a=*/false, A, /*neg_b=*/false, Bv,
        /*c_mod=*/(short)0, Cz, /*reuse_a=*/false, /*reuse_b=*/false);
#pragma unroll
    for (int v = 0; v < 8; ++v) {
      float d = D[v];
      if (d < runMin[v]) { runMin[v] = d; runIdx[v] = t + nsel; }
    }
  }

  // min+argmin across the 16 lanes of each half (rows M=v and M=v+8).
#pragma unroll
  for (int s = 1; s < 16; s <<= 1) {
#pragma unroll
    for (int v = 0; v < 8; ++v) {
      float om = __shfl_xor(runMin[v], s, 32);
      int oi = __shfl_xor(runIdx[v], s, 32);
      if (om < runMin[v] || (om == runMin[v] && oi < runIdx[v])) {
        runMin[v] = om;
        runIdx[v] = oi;
      }
    }
  }

  if ((lane & 15) == 0) {
    int rbase = m0 + ((lane >> 4) << 3);
#pragma unroll
    for (int v = 0; v < 8; ++v) {
      int row = rbase + v;
      float x = pr[row], y = pc[row];
      min_d2[(size_t)b * PN + row] = runMin[v] + (x * x + y * y);
      nn_idx[(size_t)b * PN + row] = runIdx[v];
    }
  }
}

// ----------------------------------------------- normals + dot + splat
__device__ inline void grad_at(const float* __restrict__ s, int r, int c,
                               float& gr, float& gc) {
  if (r == 0)            gr = s[WW + c] - s[c];
  else if (r == HH - 1)  gr = s[(HH - 1) * WW + c] - s[(HH - 2) * WW + c];
  else                   gr = (s[(r + 1) * WW + c] - s[(r - 1) * WW + c]) * 0.5f;
  if (c == 0)            gc = s[r * WW + 1] - s[r * WW];
  else if (c == WW - 1)  gc = s[r * WW + WW - 1] - s[r * WW + WW - 2];
  else                   gc = (s[r * WW + c + 1] - s[r * WW + c - 1]) * 0.5f;
}

__global__ __launch_bounds__(256)
void splat_kernel(const float* __restrict__ pred_all,
                  const float* __restrict__ p_r, const float* __restrict__ p_c,
                  const float* __restrict__ g_r, const float* __restrict__ g_c,
                  const float* __restrict__ min_d2, const int* __restrict__ nn_idx,
                  float* __restrict__ z_all) {
  int gid = blockIdx.x * blockDim.x + threadIdx.x;
  if (gid >= BB * PN) return;
  int b = gid / PN;
  size_t pi = (size_t)gid;
  float prr = p_r[pi], pcc = p_c[pi];
  if (prr < -1.0e3f) return;                 // invalid p sentinel (-1e4)
  float md2 = min_d2[pi];
  if (!(md2 <= 1.0f)) return;                // DIST_THRESHOLD (also rejects NaN)

  const float* sdf = pred_all + (size_t)b * HH * WW;
  int r0 = min(max((int)floorf(prr), 0), HH - 1);
  int c0 = min(max((int)floorf(pcc), 0), WW - 1);
  int r1 = min(r0 + 1, HH - 1);
  int c1 = min(c0 + 1, WW - 1);
  float dr = prr - (float)r0, dc = pcc - (float)c0;
  float w00 = (1.0f - dr) * (1.0f - dc), w01 = (1.0f - dr) * dc;
  float w10 = dr * (1.0f - dc), w11 = dr * dc;

  float g00r, g00c, g01r, g01c, g10r, g10c, g11r, g11c;
  grad_at(sdf, r0, c0, g00r, g00c);
  grad_at(sdf, r0, c1, g01r, g01c);
  grad_at(sdf, r1, c0, g10r, g10c);
  grad_at(sdf, r1, c1, g11r, g11c);
  float nr = g00r * w00 + g01r * w01 + g10r * w10 + g11r * w11;
  float nc = g00c * w00 + g01c * w01 + g10c * w10 + g11c * w11;
  // reference normalizes twice (inside _bilinear_normals and in _make_dsdf)
  float nm = sqrtf(nr * nr + nc * nc) + EPSF;
  nr /= nm; nc /= nm;
  nm = sqrtf(nr * nr + nc * nc) + EPSF;
  nr /= nm; nc /= nm;

  int j = nn_idx[pi];
  float nnr = g_r[(size_t)b * PN + j], nnc = g_c[(size_t)b * PN + j];
  float dot = ((nnr - prr) * nr + (nnc - pcc) * nc) * 100.0f;  // UPDATE_SCALE

  float* z = z_all + (size_t)b * HH * WW;
  atomicAdd(&z[r0 * WW + c0], dot * w00);
  atomicAdd(&z[r0 * WW + c1], dot * w01);
  atomicAdd(&z[r1 * WW + c0], dot * w10);
  atomicAdd(&z[r1 * WW + c1], dot * w11);
}

// ---------------------------------------------------------------- clip dsdf
__global__ void clip_kernel(float* __restrict__ z, int n) {
  int i = blockIdx.x * blockDim.x + threadIdx.x;
  if (i < n) {
    float v = z[i];
    z[i] = fminf(fmaxf(v, -1.0f), 1.0f);     // CLIP_DSDF
  }
}

// ---------------------------------------------------------------------------
extern "C" void kernel_launch(void* const* d_in, const int* in_sizes, int n_in,
                              void* d_out, int out_size, void* d_ws, size_t ws_size,
                              hipStream_t stream) {
  (void)in_sizes; (void)n_in; (void)ws_size;
  const float* pred = (const float*)d_in[0];
  const float* gt = (const float*)d_in[1];
  float* out = (float*)d_out;

  // workspace layout (floats unless noted)
  float* p_r = (float*)d_ws;               // BB*PN
  float* p_c = p_r + BB * PN;              // BB*PN
  float* g_r = p_c + BB * PN;              // BB*PN
  float* g_c = g_r + BB * PN;              // BB*PN
  float* min_d2 = g_c + BB * PN;           // BB*PN
  int* nn_idx = (int*)(min_d2 + BB * PN);  // BB*PN ints
  float* part = (float*)(nn_idx + BB * PN);// loss partials

  const int total = BB * HH * WW;

  // 1) zero output (loss slot + dsdf accumulator)
  zero_out_kernel<<<(out_size + 255) / 256, 256, 0, stream>>>(out, out_size);

  // 2) loss = mean(|pred - gt|), deterministic two-stage reduction
  const int nblk = (total + 1023) / 1024;  // 256 threads x 4 elems
  loss_partial_kernel<<<nblk, 256, 0, stream>>>(pred, gt, part, total);
  loss_final_kernel<<<1, 1024, 0, stream>>>(part, nblk, out, 1.0f / (float)total);

  // 3) ordered zero-crossing extraction (pred -> p, gt -> g)
  zc_extract_kernel<<<dim3(BB, 2), 1024, 0, stream>>>(pred, p_r, p_c, -1.0e4f);
  zc_extract_kernel<<<dim3(BB, 2), 1024, 0, stream>>>(gt, g_r, g_c, 1.0e4f);

  // 4) chamfer nearest-neighbor via V_WMMA_F32_16X16X4_F32
  nn_wmma_kernel<<<dim3(PN / 16, BB), 32, 0, stream>>>(p_r, p_c, g_r, g_c,
                                                       min_d2, nn_idx);

  // 5) normals + injected-gradient splat into d_out dsdf region
  splat_kernel<<<(BB * PN + 255) / 256, 256, 0, stream>>>(
      pred, p_r, p_c, g_r, g_c, min_d2, nn_idx, out + 1);

  // 6) clip dsdf in place
  clip_kernel<<<(total + 255) / 256, 256, 0, stream>>>(out + 1, total);
}